// OrigCapsuleRouting_58875411693956
// MI455X (gfx1250) — compile-verified
//
#include <hip/hip_runtime.h>
#include <hip/hip_bf16.h>

// ---------------------------------------------------------------------------
// Capsule routing, fully fused for MI455X (gfx1250, wave32).
//   - uhat (512MB) is NEVER materialized: recomputed per routing round from
//     L2-resident weight (32MB) + incaps (8MB) via V_WMMA_F32_16X16X4_F32.
//   - Per (b-tile=16, i-chunk=64) workgroup: WMMA computes the 16x512 uhat
//     slice for one i at a time into LDS; then wave=b / lane=o threads do the
//     b-update, wave32 shuffle-softmax over O, and c*uhat accumulation.
//   - Cross-chunk Σ_i reduction via f32 global atomics into a 256KB ws buffer;
//     squash is a tiny separate kernel between rounds.
//   - betas[r]/coups[r] regions of d_out double as storage for b and c.
// ---------------------------------------------------------------------------

#define B_   128
#define O_   32
#define I_   2048
#define DO_  16
#define DI_  8

#define BT   16           // b-tile (WMMA M)
#define CH   64           // i's per workgroup chunk
#define NBT  (B_ / BT)    // 8
#define NCH  (I_ / CH)    // 32

#define XSTRIDE 516       // incaps tile row stride (pad: 516%64=4 -> no LDS bank conflicts)
#define USTRIDE 513       // uhat/S d-major stride  (pad: 513%64=1 -> no LDS bank conflicts)

#define OUT_SZ  (B_ * O_ * DO_)          // 65536
#define RSZ     ((size_t)B_ * O_ * I_)   // 8388608 per routing-round slice

typedef __attribute__((ext_vector_type(2))) float v2f;
typedef __attribute__((ext_vector_type(8))) float v8f;

// ---------------------------------------------------------------------------
// init: coups[0] = 1/O (softmax of ones), betas[0] = 0, zero the 3 s-buffers.
// ---------------------------------------------------------------------------
__global__ void caps_init(float* __restrict__ coup0, float* __restrict__ beta0,
                          float* __restrict__ sbufs) {
    size_t idx    = (size_t)blockIdx.x * blockDim.x + threadIdx.x;
    size_t stride = (size_t)gridDim.x * blockDim.x;
    for (size_t i = idx; i < RSZ; i += stride) {
        coup0[i] = 1.0f / (float)O_;
        beta0[i] = 0.0f;
    }
    for (size_t i = idx; i < (size_t)3 * OUT_SZ; i += stride) sbufs[i] = 0.0f;
}

// ---------------------------------------------------------------------------
// One fused routing pass over all i:
//   per i: uhat[16b x (32o*16d)] via 2 chained WMMA f32 16x16x4 (K=8) -> LDS,
//   then (if !first) bn = b_prev + Σ_d outcaps*uhat -> betas_next,
//        c = softmax_o(bn) (wave32 shuffle reduce)   -> coups_next,
//   S[b,o,d] += c * uhat ; flushed with global f32 atomics at the end.
// first==1: c = 1/O, no bn/softmax (init kernel already wrote round-0 slices).
// ---------------------------------------------------------------------------
__global__ __launch_bounds__(512)
void caps_pass(const float* __restrict__ incaps,        // [B,I,Di]
               const float* __restrict__ weight,        // [O,I,Do,Di]
               const float* __restrict__ outcaps_prev,  // [B,O,Do] (unused if first)
               const float* __restrict__ beta_prev,     // [B,O,I]  (unused if first)
               float* __restrict__ beta_next,           // [B,O,I]  (unused if first)
               float* __restrict__ coup_next,           // [B,O,I]  (unused if first)
               float* __restrict__ s_accum,             // [B,O,Do] atomically accumulated
               int first)
{
    extern __shared__ float smem[];
    float* xs = smem;                   // incaps tile [16][XSTRIDE]  (b-major, i*8+k)
    float* U  = smem + 16 * XSTRIDE;    // uhat tile   [16 d][USTRIDE] -> d*USTRIDE + b*32 + o
    float* S  = U    + 16 * USTRIDE;    // s partial   same layout as U

    const int tid  = threadIdx.x;
    const int lane = tid & 31;
    const int wv   = tid >> 5;          // wave id 0..15
    const int b0   = blockIdx.x * BT;   // b-tile base
    const int i0   = blockIdx.y * CH;   // i-chunk base

    // --- load incaps tile (16 b x 64 i x 8 k = 32KB), coalesced per-b rows ---
    for (int idx = tid; idx < BT * CH * DI_; idx += 512) {
        int bb = idx >> 9;              // /512
        int r  = idx & 511;             // i*8 + k within row
        xs[bb * XSTRIDE + r] =
            incaps[(size_t)(b0 + bb) * I_ * DI_ + (size_t)i0 * DI_ + r];
    }
    for (int idx = tid; idx < 16 * USTRIDE; idx += 512) S[idx] = 0.0f;

    // phase-B mapping: b = wv, o = lane. Cache this thread's outcaps row.
    float oc[DO_];
    if (!first) {
        const float* op = outcaps_prev + ((size_t)(b0 + wv) * O_ + lane) * DO_;
#pragma unroll
        for (int d = 0; d < DO_; ++d) oc[d] = op[d];
    }

    const int hl = lane & 15;           // WMMA: M row (A) / N col=d (B,C)
    const int kb = (lane >> 4) * 2;     // WMMA: lanes 16-31 hold K=2,3 of each 4-step

    __syncthreads();

    for (int ii = 0; ii < CH; ++ii) {
        const int i = i0 + ii;

        // ---- A operands (16b x 8k) from LDS, shared by this wave's two o's ----
        // A 16x4 layout: VGPR0 = K{0|2}, VGPR1 = K{1|3} (lanes 0-15 | 16-31).
        v2f a0, a1;
        a0.x = xs[hl * XSTRIDE + ii * DI_ + kb + 0];
        a0.y = xs[hl * XSTRIDE + ii * DI_ + kb + 1];
        a1.x = xs[hl * XSTRIDE + ii * DI_ + 4 + kb + 0];
        a1.y = xs[hl * XSTRIDE + ii * DI_ + 4 + kb + 1];

#pragma unroll
        for (int oo = 0; oo < 2; ++oo) {
            const int o = wv * 2 + oo;
            // B 4x16 (K x N=d): symmetric layout, loaded straight from global
            // weight[o][i][d=hl][k], k = kb(+4*kstep) .. +1  (8B-aligned pair).
            const float* wp = weight + (((size_t)o * I_ + i) * DO_ + hl) * DI_ + kb;
            v2f bm0, bm1;
            bm0.x = wp[0]; bm0.y = wp[1];
            bm1.x = wp[4]; bm1.y = wp[5];

            v8f acc = {0.f, 0.f, 0.f, 0.f, 0.f, 0.f, 0.f, 0.f};
            acc = __builtin_amdgcn_wmma_f32_16x16x4_f32(false, a0, false, bm0,
                                                        (short)0, acc, false, false);
            acc = __builtin_amdgcn_wmma_f32_16x16x4_f32(false, a1, false, bm1,
                                                        (short)0, acc, false, false);

            // C layout: VGPR v -> M = v (lanes 0-15) / v+8 (lanes 16-31), N = lane&15 = d
            const int brow = (lane >> 4) * 8;
#pragma unroll
            for (int v = 0; v < 8; ++v)
                U[hl * USTRIDE + (brow + v) * 32 + o] = acc[v];
        }
        __syncthreads();

        // ---- phase B: thread owns (b=wv, o=lane); conflict-free LDS reads ----
        float u[DO_];
#pragma unroll
        for (int d = 0; d < DO_; ++d) u[d] = U[d * USTRIDE + wv * 32 + lane];

        float cval;
        const size_t gidx = ((size_t)(b0 + wv) * O_ + lane) * I_ + i;
        if (first) {
            cval = 1.0f / (float)O_;
        } else {
            float bn = beta_prev[gidx];
#pragma unroll
            for (int d = 0; d < DO_; ++d) bn += oc[d] * u[d];
            beta_next[gidx] = bn;
            // softmax over o = all 32 lanes of this wave (wave32)
            float m = bn;
#pragma unroll
            for (int sft = 16; sft > 0; sft >>= 1)
                m = fmaxf(m, __shfl_xor(m, sft, 32));
            float e = __expf(bn - m);
            float sum = e;
#pragma unroll
            for (int sft = 16; sft > 0; sft >>= 1)
                sum += __shfl_xor(sum, sft, 32);
            cval = e / sum;
            coup_next[gidx] = cval;
        }
#pragma unroll
        for (int d = 0; d < DO_; ++d)
            S[d * USTRIDE + wv * 32 + lane] += cval * u[d];
        __syncthreads();
    }

    // ---- flush chunk-partial s to global accumulator (32KB of f32 atomics) ----
#pragma unroll
    for (int d = 0; d < DO_; ++d)
        atomicAdd(&s_accum[((size_t)(b0 + wv) * O_ + lane) * DO_ + d],
                  S[d * USTRIDE + wv * 32 + lane]);
}

// ---------------------------------------------------------------------------
// squash over d: scale = n2/(1+n2)/(sqrt(n2)+1e-8). One thread per (b,o).
// ---------------------------------------------------------------------------
__global__ void caps_squash(const float* __restrict__ s_in, float* __restrict__ out) {
    int idx = blockIdx.x * blockDim.x + threadIdx.x;   // b*O + o
    if (idx >= B_ * O_) return;
    const float* sp = s_in + (size_t)idx * DO_;
    float v[DO_], n2 = 0.0f;
#pragma unroll
    for (int d = 0; d < DO_; ++d) { v[d] = sp[d]; n2 += v[d] * v[d]; }
    float nrm = sqrtf(n2);
    float scale = n2 / (1.0f + n2) / (nrm + 1e-8f);
    float* op = out + (size_t)idx * DO_;
#pragma unroll
    for (int d = 0; d < DO_; ++d) op[d] = scale * v[d];
}

// ---------------------------------------------------------------------------
extern "C" void kernel_launch(void* const* d_in, const int* in_sizes, int n_in,
                              void* d_out, int out_size, void* d_ws, size_t ws_size,
                              hipStream_t stream) {
    const float* incaps = (const float*)d_in[0];   // [128,2048,8]
    const float* weight = (const float*)d_in[1];   // [32,2048,16,8]

    float* out   = (float*)d_out;                  // [B,O,Do]
    float* coups = out + OUT_SZ;                   // [3,B,O,I]
    float* betas = coups + 3 * RSZ;                // [3,B,O,I]

    float* ws = (float*)d_ws;                      // ~1.25MB used
    float* s0  = ws;
    float* s1  = ws + 1 * OUT_SZ;
    float* s2  = ws + 2 * OUT_SZ;
    float* oc0 = ws + 3 * OUT_SZ;
    float* oc1 = ws + 4 * OUT_SZ;

    caps_init<<<2048, 256, 0, stream>>>(coups, betas, s0 /* s0,s1,s2 contiguous */);

    dim3 grid(NBT, NCH);
    size_t shmem = (size_t)(16 * XSTRIDE + 2 * 16 * USTRIDE) * sizeof(float); // ~98KB

    // round 0: c uniform -> s0
    caps_pass<<<grid, 512, shmem, stream>>>(incaps, weight, nullptr, nullptr,
                                            nullptr, nullptr, s0, 1);
    caps_squash<<<(B_ * O_ + 255) / 256, 256, 0, stream>>>(s0, oc0);

    // round 0->1: b1, c1 (written to betas[1]/coups[1]) and s1
    caps_pass<<<grid, 512, shmem, stream>>>(incaps, weight, oc0, betas,
                                            betas + RSZ, coups + RSZ, s1, 0);
    caps_squash<<<(B_ * O_ + 255) / 256, 256, 0, stream>>>(s1, oc1);

    // round 1->2: b2, c2 and s2
    caps_pass<<<grid, 512, shmem, stream>>>(incaps, weight, oc1, betas + RSZ,
                                            betas + 2 * RSZ, coups + 2 * RSZ, s2, 0);
    caps_squash<<<(B_ * O_ + 255) / 256, 256, 0, stream>>>(s2, out);
}